// DynamicGraphConvolution_30614526886008
// MI455X (gfx1250) — compile-verified
//
#include <hip/hip_runtime.h>
#include <math.h>

typedef float v2f __attribute__((ext_vector_type(2)));
typedef float v8f __attribute__((ext_vector_type(8)));

#define S_DIM 256
#define L_DIM 256
#define B_DIM 32
#define C_DIM 4
#define N_TOTAL  4194304u   // B*S*S*2
#define HALF_N   2097152u

__device__ __forceinline__ unsigned rotl32(unsigned x, unsigned d) {
  return (x << d) | (x >> (32u - d));
}

// Exact JAX threefry2x32 with key = jax.random.key(42) -> (k0,k1) = (0,42)
__device__ __forceinline__ uint2 threefry2x32_0_42(unsigned x0, unsigned x1) {
  const unsigned k0 = 0u, k1 = 42u;
  const unsigned k2 = k0 ^ k1 ^ 0x1BD11BDAu;
  const unsigned ra0 = 13u, ra1 = 15u, ra2 = 26u, ra3 = 6u;
  const unsigned rb0 = 17u, rb1 = 29u, rb2 = 16u, rb3 = 24u;
  x0 += k0; x1 += k1;
  x0 += x1; x1 = rotl32(x1, ra0); x1 ^= x0;
  x0 += x1; x1 = rotl32(x1, ra1); x1 ^= x0;
  x0 += x1; x1 = rotl32(x1, ra2); x1 ^= x0;
  x0 += x1; x1 = rotl32(x1, ra3); x1 ^= x0;
  x0 += k1; x1 += k2 + 1u;
  x0 += x1; x1 = rotl32(x1, rb0); x1 ^= x0;
  x0 += x1; x1 = rotl32(x1, rb1); x1 ^= x0;
  x0 += x1; x1 = rotl32(x1, rb2); x1 ^= x0;
  x0 += x1; x1 = rotl32(x1, rb3); x1 ^= x0;
  x0 += k2; x1 += k0 + 2u;
  x0 += x1; x1 = rotl32(x1, ra0); x1 ^= x0;
  x0 += x1; x1 = rotl32(x1, ra1); x1 ^= x0;
  x0 += x1; x1 = rotl32(x1, ra2); x1 ^= x0;
  x0 += x1; x1 = rotl32(x1, ra3); x1 ^= x0;
  x0 += k0; x1 += k1 + 3u;
  x0 += x1; x1 = rotl32(x1, rb0); x1 ^= x0;
  x0 += x1; x1 = rotl32(x1, rb1); x1 ^= x0;
  x0 += x1; x1 = rotl32(x1, rb2); x1 ^= x0;
  x0 += x1; x1 = rotl32(x1, rb3); x1 ^= x0;
  x0 += k1; x1 += k2 + 4u;
  x0 += x1; x1 = rotl32(x1, ra0); x1 ^= x0;
  x0 += x1; x1 = rotl32(x1, ra1); x1 ^= x0;
  x0 += x1; x1 = rotl32(x1, ra2); x1 ^= x0;
  x0 += x1; x1 = rotl32(x1, ra3); x1 ^= x0;
  x0 += k2; x1 += k0 + 5u;
  return make_uint2(x0, x1);
}

// Gumbel(-log(-log(u))) with u ~ jax.random.uniform(key(42), [B,S*S,2], 1e-10, 1.0)
__device__ __forceinline__ float gumbel_from_flat(unsigned f) {
  unsigned bits;
  if (f < HALF_N) {
    uint2 r = threefry2x32_0_42(f, f + HALF_N);
    bits = r.x;
  } else {
    uint2 r = threefry2x32_0_42(f - HALF_N, f);
    bits = r.y;
  }
  float u01 = __uint_as_float((bits >> 9) | 0x3f800000u) - 1.0f;  // [0,1)
  float u = fmaxf(1e-10f, u01 * (1.0f - 1e-10f) + 1e-10f);
  return -logf(-logf(u));
}

__device__ __forceinline__ float gelu_exact(float x) {
  return 0.5f * x * (1.0f + erff(x * 0.70710678118654752440f));
}

// ---------------------------------------------------------------------------
// Kernel 1: corr 64x16 slab per wave (4 accs, B-frag reused) -> MLP -> gumbel
//           hard argmax -> 0/1 adjacency with forced self-loops.
// ---------------------------------------------------------------------------
__global__ __launch_bounds__(256) void k_corr_adj(
    const float* __restrict__ fm,
    const float* __restrict__ W1, const float* __restrict__ b1,
    const float* __restrict__ W2, const float* __restrict__ b2,
    const float* __restrict__ W3, const float* __restrict__ b3,
    float* __restrict__ adj) {
  __shared__ float sW1[16], sb1[16], sW2[128], sb2[8], sW3[16], sb3[2];
  int tid = threadIdx.x;
  if (tid < 16)  sW1[tid] = W1[tid];
  if (tid < 16)  sb1[tid] = b1[tid];
  if (tid < 128) sW2[tid] = W2[tid];
  if (tid < 8)   sb2[tid] = b2[tid];
  if (tid < 16)  sW3[tid] = W3[tid];
  if (tid < 2)   sb3[tid] = b3[tid];
  __syncthreads();

  int wave = tid >> 5;
  int lane = tid & 31;
  int n  = lane & 15;
  int kh = lane >> 4;

  // 2048 wave-tiles = 32 b * 4 s-groups(64 rows) * 16 t-blocks
  int tile = blockIdx.x * 8 + wave;
  int b    = tile >> 6;
  int rem  = tile & 63;
  int sg   = (rem >> 4) << 6;       // s base (0,64,128,192)
  int t0   = (rem & 15) << 4;       // t base

  const float* fmb = fm + (size_t)b * S_DIM * L_DIM;
  const float* ar0 = fmb + (size_t)(sg +  0 + n) * L_DIM + 2 * kh;
  const float* ar1 = fmb + (size_t)(sg + 16 + n) * L_DIM + 2 * kh;
  const float* ar2 = fmb + (size_t)(sg + 32 + n) * L_DIM + 2 * kh;
  const float* ar3 = fmb + (size_t)(sg + 48 + n) * L_DIM + 2 * kh;
  const float* brow = fmb + (size_t)(t0 + n) * L_DIM + 2 * kh;

  v8f acc[4];
  #pragma unroll
  for (int i = 0; i < 4; ++i) acc[i] = (v8f){0.f,0.f,0.f,0.f,0.f,0.f,0.f,0.f};

  #pragma unroll 4
  for (int k0 = 0; k0 < L_DIM; k0 += 4) {
    v2f bb = *(const v2f*)(brow + k0);     // shared across 4 row-blocks
    v2f a0 = *(const v2f*)(ar0 + k0);
    v2f a1 = *(const v2f*)(ar1 + k0);
    v2f a2 = *(const v2f*)(ar2 + k0);
    v2f a3 = *(const v2f*)(ar3 + k0);
    acc[0] = __builtin_amdgcn_wmma_f32_16x16x4_f32(false, a0, false, bb, (short)0, acc[0], false, false);
    acc[1] = __builtin_amdgcn_wmma_f32_16x16x4_f32(false, a1, false, bb, (short)0, acc[1], false, false);
    acc[2] = __builtin_amdgcn_wmma_f32_16x16x4_f32(false, a2, false, bb, (short)0, acc[2], false, false);
    acc[3] = __builtin_amdgcn_wmma_f32_16x16x4_f32(false, a3, false, bb, (short)0, acc[3], false, false);
  }

  float* adjb = adj + (size_t)b * S_DIM * S_DIM;
  #pragma unroll 1
  for (int i = 0; i < 4; ++i) {
    #pragma unroll 1
    for (int v = 0; v < 8; ++v) {
      int se = sg + 16 * i + v + 8 * kh;
      int te = t0 + n;
      float x = acc[i][v];
      float h1[16];
      #pragma unroll
      for (int j = 0; j < 16; ++j) h1[j] = gelu_exact(x * sW1[j] + sb1[j]);
      float l0 = sb3[0], l1 = sb3[1];
      #pragma unroll
      for (int k = 0; k < 8; ++k) {
        float s = sb2[k];
        #pragma unroll
        for (int j = 0; j < 16; ++j) s += h1[j] * sW2[j * 8 + k];
        float h2 = gelu_exact(s);
        l0 += h2 * sW3[2 * k];
        l1 += h2 * sW3[2 * k + 1];
      }
      // log_softmax is a uniform shift -> argmax unaffected; hard gumbel
      // forward value is exactly one_hot(argmax(logits + g)).
      unsigned fbase = ((unsigned)b << 17) + (((unsigned)se << 8) + (unsigned)te) * 2u;
      float g0 = gumbel_from_flat(fbase);
      float g1 = gumbel_from_flat(fbase + 1u);
      float a0 = (l0 + g0 >= l1 + g1) ? 1.0f : 0.0f;
      if (se == te) a0 = 1.0f;               // a + (1-a)*I
      adjb[(size_t)se * S_DIM + te] = a0;
    }
  }
}

// ---------------------------------------------------------------------------
// Kernel 2: row (degree) normalization, one wave per row
// ---------------------------------------------------------------------------
__global__ __launch_bounds__(256) void k_rownorm(float* __restrict__ adj) {
  int wave = threadIdx.x >> 5, lane = threadIdx.x & 31;
  int row  = blockIdx.x * 8 + wave;          // 8192 rows
  float* r = adj + (size_t)row * S_DIM;
  float v[8];
  float s = 0.f;
  #pragma unroll
  for (int i = 0; i < 8; ++i) { v[i] = r[lane + 32 * i]; s += v[i]; }
  #pragma unroll
  for (int m = 16; m >= 1; m >>= 1) s += __shfl_xor(s, m, 32);
  float inv = 1.0f / s;
  #pragma unroll
  for (int i = 0; i < 8; ++i) r[lane + 32 * i] = v[i] * inv;
}

// ---------------------------------------------------------------------------
// Shared 64x16-slab GEMM body: D(64x16) = A(64x256) * B(256x16) [+ bias]
// A row-major ldA=256 (coalesced b64 loads), B row-major ldB=256 (strided
// column walk, fragment reused across 4 row-blocks, prefetched ahead).
// ---------------------------------------------------------------------------
__device__ __forceinline__ void gemm_slab_64x16(
    const float* __restrict__ A0, const float* __restrict__ Bp, v8f acc[4]) {
  const float* a0 = A0;
  const float* a1 = A0 + 16 * 256;
  const float* a2 = A0 + 32 * 256;
  const float* a3 = A0 + 48 * 256;
  #pragma unroll 4
  for (int k0 = 0; k0 < 256; k0 += 4) {
    if (k0 + 16 < 256)
      __builtin_prefetch(Bp + (size_t)(k0 + 16) * 256, 0, 1);  // global_prefetch
    v2f bb;
    bb.x = Bp[(size_t)k0 * 256];
    bb.y = Bp[(size_t)(k0 + 1) * 256];
    v2f va0 = *(const v2f*)(a0 + k0);
    v2f va1 = *(const v2f*)(a1 + k0);
    v2f va2 = *(const v2f*)(a2 + k0);
    v2f va3 = *(const v2f*)(a3 + k0);
    acc[0] = __builtin_amdgcn_wmma_f32_16x16x4_f32(false, va0, false, bb, (short)0, acc[0], false, false);
    acc[1] = __builtin_amdgcn_wmma_f32_16x16x4_f32(false, va1, false, bb, (short)0, acc[1], false, false);
    acc[2] = __builtin_amdgcn_wmma_f32_16x16x4_f32(false, va2, false, bb, (short)0, acc[2], false, false);
    acc[3] = __builtin_amdgcn_wmma_f32_16x16x4_f32(false, va3, false, bb, (short)0, acc[3], false, false);
  }
}

// Kernel 3: G = F @ Wl + bl  (bias fused here; adj rows sum to 1 after norm)
__global__ __launch_bounds__(256) void k_lin(
    const float* __restrict__ F, const float* __restrict__ Wl,
    const float* __restrict__ bl, float* __restrict__ G) {
  int wave = threadIdx.x >> 5, lane = threadIdx.x & 31;
  int n = lane & 15, kh = lane >> 4;
  // 8192 wave-tiles = 128 bc * 4 row-groups(64) * 16 col-blocks
  int tile = blockIdx.x * 8 + wave;
  int bc   = tile >> 6;
  int rem  = tile & 63;
  int r0   = (rem >> 4) << 6;
  int c0   = (rem & 15) << 4;
  const float* A0 = F + (size_t)bc * S_DIM * 256 + (size_t)(r0 + n) * 256 + 2 * kh;
  const float* Bp = Wl + (size_t)(2 * kh) * 256 + (c0 + n);
  v8f acc[4];
  #pragma unroll
  for (int i = 0; i < 4; ++i) acc[i] = (v8f){0.f,0.f,0.f,0.f,0.f,0.f,0.f,0.f};
  gemm_slab_64x16(A0, Bp, acc);
  float bias = bl[c0 + n];
  float* Gt = G + (size_t)bc * S_DIM * 256;
  #pragma unroll
  for (int i = 0; i < 4; ++i)
    #pragma unroll
    for (int v = 0; v < 8; ++v)
      Gt[(size_t)(r0 + 16 * i + v + 8 * kh) * 256 + c0 + n] = acc[i][v] + bias;
}

// Kernel 4: out = adj_n @ G
__global__ __launch_bounds__(256) void k_prop(
    const float* __restrict__ adj, const float* __restrict__ G,
    float* __restrict__ out) {
  int wave = threadIdx.x >> 5, lane = threadIdx.x & 31;
  int n = lane & 15, kh = lane >> 4;
  int tile = blockIdx.x * 8 + wave;          // 8192 wave-tiles
  int bc   = tile >> 6;
  int b    = bc >> 2;
  int rem  = tile & 63;
  int r0   = (rem >> 4) << 6;
  int c0   = (rem & 15) << 4;
  const float* A0 = adj + (size_t)b * S_DIM * S_DIM + (size_t)(r0 + n) * S_DIM + 2 * kh;
  const float* Bp = G + (size_t)bc * S_DIM * 256 + (size_t)(2 * kh) * 256 + (c0 + n);
  v8f acc[4];
  #pragma unroll
  for (int i = 0; i < 4; ++i) acc[i] = (v8f){0.f,0.f,0.f,0.f,0.f,0.f,0.f,0.f};
  gemm_slab_64x16(A0, Bp, acc);
  float* Ot = out + (size_t)bc * S_DIM * 256;
  #pragma unroll
  for (int i = 0; i < 4; ++i)
    #pragma unroll
    for (int v = 0; v < 8; ++v)
      Ot[(size_t)(r0 + 16 * i + v + 8 * kh) * 256 + c0 + n] = acc[i][v];
}

extern "C" void kernel_launch(void* const* d_in, const int* in_sizes, int n_in,
                              void* d_out, int out_size, void* d_ws, size_t ws_size,
                              hipStream_t stream) {
  const float* features = (const float*)d_in[0];  // [B,C,S,Din]
  const float* fm       = (const float*)d_in[1];  // [B,S,L]
  const float* W1 = (const float*)d_in[2];
  const float* b1 = (const float*)d_in[3];
  const float* W2 = (const float*)d_in[4];
  const float* b2 = (const float*)d_in[5];
  const float* W3 = (const float*)d_in[6];
  const float* b3 = (const float*)d_in[7];
  const float* Wl = (const float*)d_in[8];
  const float* bl = (const float*)d_in[9];
  float* out = (float*)d_out;

  float* adj = (float*)d_ws;                                   // 8 MB  [B,S,S]
  float* G   = (float*)((char*)d_ws +
                        (size_t)B_DIM * S_DIM * S_DIM * sizeof(float)); // 33.5 MB

  k_corr_adj<<< 256, 256, 0, stream>>>(fm, W1, b1, W2, b2, W3, b3, adj);
  k_rownorm <<<1024, 256, 0, stream>>>(adj);
  k_lin     <<<1024, 256, 0, stream>>>(features, Wl, bl, G);
  k_prop    <<<1024, 256, 0, stream>>>(adj, G, out);
}